// MultiHeadedAttention_18287970746520
// MI455X (gfx1250) — compile-verified
//
#include <hip/hip_runtime.h>
#include <cmath>

typedef __attribute__((ext_vector_type(2))) float v2f;
typedef __attribute__((ext_vector_type(8))) float v8f;

static constexpr int kS     = 1600;  // H*W
static constexpr int kC     = 128;   // channels / EMBED
static constexpr int kNB    = 8;     // batch
static constexpr int kHEADS = 8;
static constexpr int kHD    = 16;    // head dim

// D = A(16x4 f32) * B(4x16 f32) + C(16x16 f32), fp32 WMMA
__device__ __forceinline__ v8f wmma_f32(v2f a, v2f b, v8f c) {
    // 8 args: (neg_a, A, neg_b, B, c_mod, C, reuse_a, reuse_b)
    return __builtin_amdgcn_wmma_f32_16x16x4_f32(false, a, false, b, (short)0, c,
                                                 false, false);
}

__device__ __forceinline__ float xor16(float x) { return __shfl_xor(x, 16, 32); }

// ---------------------------------------------------------------------------
// Kernel 1: QKV projection.  qkv[n,o,s] = sum_c w_qkv[o,c] * x[n,c,s]
// o = head*48 + which*16 + d  -> scatter into Q/K/V[n,head,d,s].
// Q is pre-scaled by 1/sqrt(HD) = 0.25 so attention needs no extra scaling.
// One wave computes a 16(o) x 16(s) tile, K loop over c in steps of 4.
// ---------------------------------------------------------------------------
__global__ void qkv_proj_kernel(const float* __restrict__ x,
                                const float* __restrict__ w_qkv,
                                float* __restrict__ Qb, float* __restrict__ Kb,
                                float* __restrict__ Vb) {
    const int ot   = blockIdx.x;          // 0..23 : o-tile = head*3 + which
    const int st   = blockIdx.y;          // 0..99 : s-tile
    const int nb   = blockIdx.z;          // 0..7
    const int lane = threadIdx.x;         // 0..31
    const int h    = lane >> 4;           // lane half
    const int li   = lane & 15;

    const int head  = ot / 3;
    const int which = ot % 3;
    const int s     = st * 16 + li;

    const float* wrow  = w_qkv + (size_t)(ot * 16 + li) * kC;   // A row m=li
    const float* xbase = x + (size_t)nb * kC * kS + s;          // B col n=s

    v8f acc = {};
#pragma unroll 8
    for (int c0 = 0; c0 < kC; c0 += 4) {
        v2f a, b;
        a.x = wrow[c0 + 2 * h + 0];
        a.y = wrow[c0 + 2 * h + 1];
        b.x = xbase[(size_t)(c0 + 2 * h + 0) * kS];
        b.y = xbase[(size_t)(c0 + 2 * h + 1) * kS];
        acc = wmma_f32(a, b, acc);
    }

    float* dst = (which == 0) ? Qb : (which == 1) ? Kb : Vb;
    dst += (size_t)(nb * kHEADS + head) * kHD * kS;
    const float mul = (which == 0) ? 0.25f : 1.0f;   // fold 1/sqrt(16) into Q
#pragma unroll
    for (int r = 0; r < 8; ++r) {
        const int d = r + 8 * h;                      // D row M = r + 8*half
        dst[(size_t)d * kS + s] = acc[r] * mul;
    }
}

// ---------------------------------------------------------------------------
// Kernel 2: flash attention per (n, head, i-tile).  One wave per 16-row i-tile.
// E^T tile (j x i) = A(K-tile) * B(Q-tile); online softmax over j (M dim);
// ctx accumulates V * P with P re-shuffled from C/D layout to B layout.
// ---------------------------------------------------------------------------
__global__ void attn_kernel(const float* __restrict__ Qb,
                            const float* __restrict__ Kb,
                            const float* __restrict__ Vb,
                            float* __restrict__ CTX) {
    const int it   = blockIdx.x;          // 0..99 : i-tile
    const int head = blockIdx.y;
    const int nb   = blockIdx.z;
    const int lane = threadIdx.x;
    const int h    = lane >> 4;
    const int li   = lane & 15;

    const size_t base = (size_t)(nb * kHEADS + head) * kHD * kS;
    const float* q = Qb + base;
    const float* k = Kb + base;
    const float* v = Vb + base;

    const int i = it * 16 + li;           // this lane's column (query index)

    // Preload B(Q) operand for all 4 K-steps: B[c][i], c = 4*kk + 2h + {0,1}
    float bq[8];
#pragma unroll
    for (int kk = 0; kk < 4; ++kk) {
        bq[2 * kk + 0] = q[(size_t)(4 * kk + 2 * h + 0) * kS + i];
        bq[2 * kk + 1] = q[(size_t)(4 * kk + 2 * h + 1) * kS + i];
    }

    float m_run = -INFINITY;
    float l_run = 0.0f;
    v8f   acc   = {};

    for (int jt = 0; jt < 100; ++jt) {
        const int j0 = jt * 16;

        // E^T tile: rows j, cols i.  A = K-tile (M=j, K=c), B = Q (preloaded).
        v8f e = {};
#pragma unroll
        for (int kk = 0; kk < 4; ++kk) {
            v2f a, b;
            a.x = k[(size_t)(4 * kk + 2 * h + 0) * kS + (j0 + li)];
            a.y = k[(size_t)(4 * kk + 2 * h + 1) * kS + (j0 + li)];
            b.x = bq[2 * kk + 0];
            b.y = bq[2 * kk + 1];
            e = wmma_f32(a, b, e);
        }

        // Online softmax over j. Lane holds rows j = r + 8h for column i; the
        // two lane-halves exchange partial stats so both carry identical m/l.
        float pmax = e[0];
#pragma unroll
        for (int r = 1; r < 8; ++r) pmax = fmaxf(pmax, e[r]);
        pmax = fmaxf(pmax, xor16(pmax));
        const float m_new = fmaxf(m_run, pmax);
        const float alpha = __expf(m_run - m_new);
        float p[8], psum = 0.0f;
#pragma unroll
        for (int r = 0; r < 8; ++r) {
            p[r] = __expf(e[r] - m_new);
            psum += p[r];
        }
        psum += xor16(psum);
        l_run = l_run * alpha + psum;
        m_run = m_new;
#pragma unroll
        for (int r = 0; r < 8; ++r) acc[r] *= alpha;

        // Re-shape P (C/D layout: reg r = row j=r+8h, col i) into B-operand
        // layout: step kk needs P[4kk + 2h + {0,1}][i] -> 8 xor-16 shuffles.
        const float s0 = xor16(p[0]), s1 = xor16(p[1]);
        const float s2 = xor16(p[2]), s3 = xor16(p[3]);
        const float s4 = xor16(p[4]), s5 = xor16(p[5]);
        const float s6 = xor16(p[6]), s7 = xor16(p[7]);

        v2f bp[4];
        bp[0].x = h ? s2 : p[0];  bp[0].y = h ? s3 : p[1];   // j = 0,1 | 2,3
        bp[1].x = h ? s6 : p[4];  bp[1].y = h ? s7 : p[5];   // j = 4,5 | 6,7
        bp[2].x = h ? p[2] : s0;  bp[2].y = h ? p[3] : s1;   // j = 8,9 |10,11
        bp[3].x = h ? p[6] : s4;  bp[3].y = h ? p[7] : s5;   // j =12,13|14,15

        // acc[c][i] += sum_j V[c][j] * P[j][i].  A = V-tile (M=c, K=j).
#pragma unroll
        for (int kk = 0; kk < 4; ++kk) {
            v2f a;
            a.x = v[(size_t)li * kS + (j0 + 4 * kk + 2 * h + 0)];
            a.y = v[(size_t)li * kS + (j0 + 4 * kk + 2 * h + 1)];
            acc = wmma_f32(a, bp[kk], acc);
        }
    }

    const float inv = 1.0f / l_run;       // per-column normalizer (same in both halves)
    float* ctx = CTX + base;
#pragma unroll
    for (int r = 0; r < 8; ++r) {
        const int c = r + 8 * h;
        ctx[(size_t)c * kS + i] = acc[r] * inv;
    }
}

// ---------------------------------------------------------------------------
// Kernel 3: output projection.  out[n,oc,s] = sum_c w_o[oc,c]*ctx[n,c,s] + b_o[oc]
// ctx buffer is [n][head*16+d][s] == [n][c][s] contiguous.
// ---------------------------------------------------------------------------
__global__ void out_proj_kernel(const float* __restrict__ CTX,
                                const float* __restrict__ w_o,
                                const float* __restrict__ b_o,
                                float* __restrict__ out) {
    const int ot   = blockIdx.x;          // 0..7  : oc-tile
    const int st   = blockIdx.y;          // 0..99 : s-tile
    const int nb   = blockIdx.z;
    const int lane = threadIdx.x;
    const int h    = lane >> 4;
    const int li   = lane & 15;
    const int s    = st * 16 + li;

    const float* wrow  = w_o + (size_t)(ot * 16 + li) * kC;
    const float* cbase = CTX + (size_t)nb * kC * kS + s;

    v8f acc = {};
#pragma unroll 8
    for (int c0 = 0; c0 < kC; c0 += 4) {
        v2f a, b;
        a.x = wrow[c0 + 2 * h + 0];
        a.y = wrow[c0 + 2 * h + 1];
        b.x = cbase[(size_t)(c0 + 2 * h + 0) * kS];
        b.y = cbase[(size_t)(c0 + 2 * h + 1) * kS];
        acc = wmma_f32(a, b, acc);
    }
#pragma unroll
    for (int r = 0; r < 8; ++r) {
        const int oc = ot * 16 + r + 8 * h;
        out[((size_t)nb * kC + oc) * kS + s] = acc[r] + b_o[oc];
    }
}

// ---------------------------------------------------------------------------
extern "C" void kernel_launch(void* const* d_in, const int* in_sizes, int n_in,
                              void* d_out, int out_size, void* d_ws, size_t ws_size,
                              hipStream_t stream) {
    const float* x     = (const float*)d_in[0];   // (8,128,40,40)
    const float* w_qkv = (const float*)d_in[1];   // (384,128)
    const float* w_o   = (const float*)d_in[2];   // (128,128)
    const float* b_o   = (const float*)d_in[3];   // (128,)
    float*       out   = (float*)d_out;           // (8,128,40,40)

    const size_t per = (size_t)kNB * kC * kS;     // 1,638,400 floats (6.55 MB)
    float* Qb  = (float*)d_ws;                    // pre-scaled by 1/sqrt(HD)
    float* Kb  = Qb + per;
    float* Vb  = Kb + per;
    float* CTX = Vb + per;                        // total ws: ~26.2 MB

    qkv_proj_kernel<<<dim3(24, 100, kNB), 32, 0, stream>>>(x, w_qkv, Qb, Kb, Vb);
    attn_kernel<<<dim3(100, kHEADS, kNB), 32, 0, stream>>>(Qb, Kb, Vb, CTX);
    out_proj_kernel<<<dim3(8, 100, kNB), 32, 0, stream>>>(CTX, w_o, b_o, out);
}